// CLAPP_layer_89008902242561
// MI455X (gfx1250) — compile-verified
//
#include <hip/hip_runtime.h>

// ---------------------------------------------------------------------------
// CLAPP layer on MI455X (gfx1250, wave32).
// H = I = 8192. Entirely HBM-bandwidth bound (~1.25 GB traffic).
// dW / dW_pred tiles are computed with V_WMMA_F32_16X16X4_F32 (rank-2 /
// rank-1 updates expressed as 16x4 @ 4x16 matmuls with C = 0).
// ---------------------------------------------------------------------------

#define HD   8192                 // hidden size
#define NI   8192                 // input size
#define PI_F 3.14159265358979323846f

typedef float v2f __attribute__((ext_vector_type(2)));
typedef float v8f __attribute__((ext_vector_type(8)));

__device__ __forceinline__ float surrogate(float x) {
    float px = PI_F * x;
    return 1.0f / (PI_F * (1.0f + px * px));
}

__device__ __forceinline__ float wave_reduce_add(float s) {
    #pragma unroll
    for (int off = 16; off >= 1; off >>= 1)
        s += __shfl_xor(s, off, 32);
    return s;
}

// ---------------------------------------------------------------------------
// K1: cur = W_fc @ inp, fused Leaky step. One wave per output row.
// Lane l reads cols [4l, 4l+3] stepping 128 -> 512B contiguous per wave/iter.
// ---------------------------------------------------------------------------
__global__ void k1_fc_leaky(const float* __restrict__ W, const float* __restrict__ x,
                            const float* __restrict__ mem,
                            float* __restrict__ spk_out, float* __restrict__ mem_out) {
    const int wave = threadIdx.x >> 5;
    const int lane = threadIdx.x & 31;
    const int row  = blockIdx.x * 8 + wave;
    const float* wr = W + (size_t)row * NI;

    float s = 0.0f;
    for (int k = lane * 4; k < NI; k += 128) {
        float4 w4 = *(const float4*)(wr + k);
        float4 x4 = *(const float4*)(x + k);
        s = fmaf(w4.x, x4.x, s);
        s = fmaf(w4.y, x4.y, s);
        s = fmaf(w4.z, x4.z, s);
        s = fmaf(w4.w, x4.w, s);
    }
    s = wave_reduce_add(s);
    if (lane == 0) {
        float nm  = fmaf(0.9f, mem[row], s);     // BETA * mem + cur
        float spk = (nm > 1.0f) ? 1.0f : 0.0f;   // fire
        spk_out[row] = spk;
        mem_out[row] = nm - spk;                 // reset by subtraction (THR=1)
    }
}

// ---------------------------------------------------------------------------
// K2: feedback_new = W_pred @ spk. Same wave-per-row GEMV.
// ---------------------------------------------------------------------------
__global__ void k2_fbnew(const float* __restrict__ W, const float* __restrict__ spk,
                         float* __restrict__ fb_out) {
    const int wave = threadIdx.x >> 5;
    const int lane = threadIdx.x & 31;
    const int row  = blockIdx.x * 8 + wave;
    const float* wr = W + (size_t)row * HD;

    float s = 0.0f;
    for (int k = lane * 4; k < HD; k += 128) {
        float4 w4 = *(const float4*)(wr + k);
        float4 x4 = *(const float4*)(spk + k);
        s = fmaf(w4.x, x4.x, s);
        s = fmaf(w4.y, x4.y, s);
        s = fmaf(w4.z, x4.z, s);
        s = fmaf(w4.w, x4.w, s);
    }
    s = wave_reduce_add(s);
    if (lane == 0) fb_out[row] = s;
}

// ---------------------------------------------------------------------------
// K2b: retrodiction partials. retro[j] = sum_h spk[h] * W_pred[h][j].
// 16 row-blocks x 32 col-blocks; thread-per-column -> fully coalesced rows.
// Deterministic (fixed reduction tree, no atomics).
// ---------------------------------------------------------------------------
__global__ void k2b_retro_part(const float* __restrict__ W, const float* __restrict__ spk,
                               float* __restrict__ part) {
    const int cb  = blockIdx.x & 31;           // 32 column blocks of 256
    const int rb  = blockIdx.x >> 5;           // 16 row blocks of 512
    const int col = cb * 256 + threadIdx.x;
    const float* wp = W + (size_t)(rb * 512) * HD + col;
    const float* sp = spk + rb * 512;

    float s = 0.0f;
    for (int h = 0; h < 512; ++h)
        s = fmaf(sp[h], wp[(size_t)h * HD], s);
    part[rb * HD + col] = s;
}

__global__ void k2c_retro_reduce(const float* __restrict__ part, float* __restrict__ retro) {
    const int col = blockIdx.x * 256 + threadIdx.x;
    float s = 0.0f;
    #pragma unroll
    for (int rb = 0; rb < 16; ++rb)
        s += part[rb * HD + col];
    retro[col] = s;
}

// ---------------------------------------------------------------------------
// K3a: per-row gradient coefficients (bf folded in).
//   a[h]  = bf * feedback[h] * surr(mem_out[h])
//   b[h]  = bf * retro[h]    * surr(prev_mem[h])
//   sb[h] = bf * spk[h]
// ---------------------------------------------------------------------------
__global__ void k3a_coefs(const int* __restrict__ bf, const float* __restrict__ feedback,
                          const float* __restrict__ mem_out, const float* __restrict__ retro,
                          const float* __restrict__ prev_mem, const float* __restrict__ spk,
                          float* __restrict__ a, float* __restrict__ b, float* __restrict__ sb) {
    const int h = blockIdx.x * 256 + threadIdx.x;
    const float bfv = (float)bf[0];
    a[h]  = bfv * feedback[h] * surrogate(mem_out[h]);
    b[h]  = bfv * retro[h]    * surrogate(prev_mem[h]);
    sb[h] = bfv * spk[h];
}

// ---------------------------------------------------------------------------
// K3b: dW and dW_pred 16x16 tiles via V_WMMA_F32_16X16X4_F32.
//   dW tile      = A(16x4){a,b,0,0}      x  B(4x16){inp, prev_inp, 0, 0}
//   dW_pred tile = A(16x4){bf*spk,0,0,0} x  B(4x16){prev_spk, 0, 0, 0}
// A layout (ISA 7.12.2): lanes 0-15 hold K=0 (v.x) and K=1 (v.y) for M=lane;
// lanes 16-31 hold K=2,3 (zero here). B layout: VGPR0 lanes 0-15 = row K=0,
// VGPR1 lanes 0-15 = row K=1; upper lanes = rows K=2,3 (zero here).
// D layout: component r -> row m0 + r + 8*(lane>=16), col n0 + (lane&15).
// One wave per tile, 8 waves per block.
// ---------------------------------------------------------------------------
__global__ void k3b_outer_wmma(const float* __restrict__ a, const float* __restrict__ b,
                               const float* __restrict__ sb,
                               const float* __restrict__ inp, const float* __restrict__ pinp,
                               const float* __restrict__ pspk,
                               float* __restrict__ dW, float* __restrict__ dWp) {
    const int wave   = threadIdx.x >> 5;
    const int lane   = threadIdx.x & 31;
    const int tileId = blockIdx.x * 8 + wave;          // 512*512 tiles
    const int tn = tileId & 511;                       // tile col index
    const int tm = tileId >> 9;                        // tile row index
    const int m0 = tm * 16;
    const int n0 = tn * 16;
    const int l16  = lane & 15;
    const bool lo  = (lane < 16);

    // Operands (upper half-lanes carry K=2,3 == 0)
    float av = a[m0 + l16], bv = b[m0 + l16];
    float xv = inp[n0 + l16], pv = pinp[n0 + l16];
    float sv = sb[m0 + l16], qv = pspk[n0 + l16];

    v2f A1;  A1.x = lo ? av : 0.0f;  A1.y = lo ? bv : 0.0f;
    v2f B1;  B1.x = lo ? xv : 0.0f;  B1.y = lo ? pv : 0.0f;
    v2f A2;  A2.x = lo ? sv : 0.0f;  A2.y = 0.0f;
    v2f B2;  B2.x = lo ? qv : 0.0f;  B2.y = 0.0f;
    v8f c = {};

    v8f d1 = __builtin_amdgcn_wmma_f32_16x16x4_f32(false, A1, false, B1,
                                                   (short)0, c, false, false);
    v8f d2 = __builtin_amdgcn_wmma_f32_16x16x4_f32(false, A2, false, B2,
                                                   (short)0, c, false, false);

    const int rowOff = (lane >> 4) * 8;                // +8 rows for upper half-lanes
    const size_t colIdx = (size_t)(n0 + l16);
    #pragma unroll
    for (int r = 0; r < 8; ++r) {
        size_t idx = (size_t)(m0 + r + rowOff) * NI + colIdx;
        dW[idx]  = d1[r];
        dWp[idx] = d2[r];
    }
}

// ---------------------------------------------------------------------------
// K4: loss = -bf * <spk, feedback> / H. Single block, deterministic tree.
// ---------------------------------------------------------------------------
__global__ void k4_loss(const int* __restrict__ bf, const float* __restrict__ spk,
                        const float* __restrict__ feedback, float* __restrict__ loss) {
    __shared__ float red[8];
    const int wave = threadIdx.x >> 5;
    const int lane = threadIdx.x & 31;
    float s = 0.0f;
    for (int i = threadIdx.x; i < HD; i += 256)
        s = fmaf(spk[i], feedback[i], s);
    s = wave_reduce_add(s);
    if (lane == 0) red[wave] = s;
    __syncthreads();
    if (threadIdx.x == 0) {
        float t = 0.0f;
        #pragma unroll
        for (int w = 0; w < 8; ++w) t += red[w];
        loss[0] = -((float)bf[0]) * t / (float)HD;
    }
}

// ---------------------------------------------------------------------------
extern "C" void kernel_launch(void* const* d_in, const int* in_sizes, int n_in,
                              void* d_out, int out_size, void* d_ws, size_t ws_size,
                              hipStream_t stream) {
    const float* inp      = (const float*)d_in[0];
    const int*   bf       = (const int*)  d_in[1];
    const float* mem      = (const float*)d_in[2];
    const float* prev_mem = (const float*)d_in[3];
    const float* prev_inp = (const float*)d_in[4];
    const float* prev_spk = (const float*)d_in[5];
    const float* feedback = (const float*)d_in[6];
    const float* W_fc     = (const float*)d_in[7];
    const float* W_pred   = (const float*)d_in[8];

    // Output layout: spk[H], mem_out[H], loss[1], dW[H*I], dW_pred[H*H], fb_new[H]
    float* out      = (float*)d_out;
    float* o_spk    = out;
    float* o_mem    = out + HD;
    float* o_loss   = out + 2 * HD;
    float* o_dW     = out + 2 * HD + 1;
    float* o_dWp    = o_dW + (size_t)HD * NI;
    float* o_fbnew  = o_dWp + (size_t)HD * HD;

    // Workspace layout (floats): retro partials [16][H], retro [H], a [H], b [H], sb [H]
    float* wsf      = (float*)d_ws;
    float* ws_part  = wsf;
    float* ws_retro = wsf + 16 * HD;
    float* ws_a     = ws_retro + HD;
    float* ws_b     = ws_a + HD;
    float* ws_sb    = ws_b + HD;

    // K1: cur = W_fc @ inp + LIF  -> spk, mem_out
    k1_fc_leaky<<<HD / 8, 256, 0, stream>>>(W_fc, inp, mem, o_spk, o_mem);

    // K2: feedback_new = W_pred @ spk
    k2_fbnew<<<HD / 8, 256, 0, stream>>>(W_pred, o_spk, o_fbnew);

    // K2b/K2c: retrodiction = spk @ W_pred (deterministic two-pass reduction)
    k2b_retro_part<<<16 * 32, 256, 0, stream>>>(W_pred, o_spk, ws_part);
    k2c_retro_reduce<<<HD / 256, 256, 0, stream>>>(ws_part, ws_retro);

    // K3a: per-row coefficients (bf folded in)
    k3a_coefs<<<HD / 256, 256, 0, stream>>>(bf, feedback, o_mem, ws_retro,
                                            prev_mem, o_spk, ws_a, ws_b, ws_sb);

    // K3b: dW and dW_pred tiles via WMMA (512*512 tiles, 8 per block)
    k3b_outer_wmma<<<(512 * 512) / 8, 256, 0, stream>>>(ws_a, ws_b, ws_sb,
                                                        inp, prev_inp, prev_spk,
                                                        o_dW, o_dWp);

    // K4: loss
    k4_loss<<<1, 256, 0, stream>>>(bf, o_spk, feedback, o_loss);
}